// FusionLightGCNModel_13503377179002
// MI455X (gfx1250) — compile-verified
//
#include <hip/hip_runtime.h>

typedef float v2f __attribute__((ext_vector_type(2)));
typedef float v8f __attribute__((ext_vector_type(8)));

#define EMBED 64
#define EMBED4 16  // EMBED/4

// ---------------- utility kernels ----------------

__global__ void fill_zero4_kernel(float4* __restrict__ p, long n4) {
    long i = (long)blockIdx.x * blockDim.x + threadIdx.x;
    if (i < n4) p[i] = make_float4(0.f, 0.f, 0.f, 0.f);
}

__global__ void init_copy_kernel(const float4* __restrict__ src,
                                 float4* __restrict__ cur,
                                 float4* __restrict__ res, long n4) {
    long i = (long)blockIdx.x * blockDim.x + threadIdx.x;
    if (i < n4) {
        float4 v = src[i];
        cur[i] = v;
        res[i] = v;
    }
}

__global__ void accum_kernel(float4* __restrict__ res,
                             const float4* __restrict__ nw, long n4) {
    long i = (long)blockIdx.x * blockDim.x + threadIdx.x;
    if (i < n4) {
        float4 r = res[i];
        float4 a = nw[i];
        r.x += a.x; r.y += a.y; r.z += a.z; r.w += a.w;
        res[i] = r;
    }
}

// ---------------- degrees ----------------

__global__ void degree2_kernel(const int* __restrict__ ia,
                               const int* __restrict__ ib,
                               float* __restrict__ da,
                               float* __restrict__ db, int E) {
    int e = blockIdx.x * blockDim.x + threadIdx.x;
    if (e < E) {
        unsafeAtomicAdd(&da[ia[e]], 1.0f);
        unsafeAtomicAdd(&db[ib[e]], 1.0f);
    }
}

// p = rsqrt(max(p,1)) over the whole (padded) degree region
__global__ void rsqrt_kernel(float* __restrict__ p, long n) {
    long i = (long)blockIdx.x * blockDim.x + threadIdx.x;
    if (i < n) p[i] = rsqrtf(fmaxf(p[i], 1.0f));
}

// ---------------- graph conv (scatter-add) ----------------
// 16 lanes cooperate on one edge; lane sub handles dims [4*sub, 4*sub+4).
// Both directions of the rate relation share the same weight r_u[u]*r_i[i].
__global__ void conv_rate_kernel(const float4* __restrict__ cur_u,
                                 const float4* __restrict__ cur_i,
                                 float* __restrict__ new_u,
                                 float* __restrict__ new_i,
                                 const int* __restrict__ ru,
                                 const int* __restrict__ ri,
                                 const float* __restrict__ r_u,
                                 const float* __restrict__ r_i, int E) {
    long t = (long)blockIdx.x * blockDim.x + threadIdx.x;
    int e = (int)(t >> 4);
    int sub = (int)(t & 15);
    if (e >= E) return;
    int u = ru[e], i = ri[e];
    float w = r_u[u] * r_i[i];
    float4 hi = cur_i[(size_t)i * EMBED4 + sub];   // item row (src for i->u)
    float4 hu = cur_u[(size_t)u * EMBED4 + sub];   // user row (src for u->i)
    float* pu = new_u + (size_t)u * EMBED + sub * 4;
    float* pi = new_i + (size_t)i * EMBED + sub * 4;
    unsafeAtomicAdd(pu + 0, hi.x * w);
    unsafeAtomicAdd(pu + 1, hi.y * w);
    unsafeAtomicAdd(pu + 2, hi.z * w);
    unsafeAtomicAdd(pu + 3, hi.w * w);
    unsafeAtomicAdd(pi + 0, hu.x * w);
    unsafeAtomicAdd(pi + 1, hu.y * w);
    unsafeAtomicAdd(pi + 2, hu.z * w);
    unsafeAtomicAdd(pi + 3, hu.w * w);
}

__global__ void conv_trust_kernel(const float4* __restrict__ cur_u,
                                  float* __restrict__ new_u,
                                  const int* __restrict__ src,
                                  const int* __restrict__ dst,
                                  const float* __restrict__ r_ts,
                                  const float* __restrict__ r_td, int E) {
    long t = (long)blockIdx.x * blockDim.x + threadIdx.x;
    int e = (int)(t >> 4);
    int sub = (int)(t & 15);
    if (e >= E) return;
    int s = src[e], d = dst[e];
    float w = r_ts[s] * r_td[d];
    float4 hs = cur_u[(size_t)s * EMBED4 + sub];
    float* pd = new_u + (size_t)d * EMBED + sub * 4;
    unsafeAtomicAdd(pd + 0, hs.x * w);
    unsafeAtomicAdd(pd + 1, hs.y * w);
    unsafeAtomicAdd(pd + 2, hs.z * w);
    unsafeAtomicAdd(pd + 3, hs.w * w);
}

// ---------------- scoring via f32 WMMA ----------------
// One wave handles 16 edges: A[m][k] = res_u[iu[m]][k], B[k][n] = res_i[ii[n]][k].
// D = A*B (16x16x4 WMMA, 16 K-steps); diagonal D[m][m] = dot product m.
// Tiles [0, tilesPerSide) -> pos scores, [tilesPerSide, 2*tilesPerSide) -> neg.
__global__ void score_wmma_kernel(const float* __restrict__ res_u,
                                  const float* __restrict__ res_i,
                                  const int* __restrict__ pos_u,
                                  const int* __restrict__ pos_i,
                                  const int* __restrict__ neg_u,
                                  const int* __restrict__ neg_i,
                                  float* __restrict__ out,
                                  int nPred, int tilesPerSide) {
    int wave = threadIdx.x >> 5;
    int lane = threadIdx.x & 31;
    int tile = blockIdx.x * (blockDim.x >> 5) + wave;
    if (tile >= 2 * tilesPerSide) return;     // uniform per wave

    const int* iu;
    const int* ii;
    float* obase;
    int t = tile;
    if (t < tilesPerSide) { iu = pos_u; ii = pos_i; obase = out; }
    else { t -= tilesPerSide; iu = neg_u; ii = neg_i; obase = out + nPred; }

    int tbase = t * 16;
    int m = lane & 15;
    int em = tbase + m;
    if (em >= nPred) em = nPred - 1;          // clamp (writes guarded below)
    int eu = iu[em];
    int ei = ii[em];
    const v2f* rowU = (const v2f*)(res_u + (size_t)eu * EMBED);
    const v2f* rowI = (const v2f*)(res_i + (size_t)ei * EMBED);
    int half = lane >> 4;                      // lanes 0-15: K pair 0, 16-31: K pair 1

    v8f c = {};
#pragma unroll
    for (int s = 0; s < 16; ++s) {
        // A layout (16x4 f32): lane L<16 holds M=L, K = 4s,4s+1; L>=16 holds K = 4s+2,4s+3
        v2f a = rowU[2 * s + half];
        // B layout (4x16 f32): lane L<16 holds N=L, K = 4s,4s+1; L>=16 holds K = 4s+2,4s+3
        v2f b = rowI[2 * s + half];
        c = __builtin_amdgcn_wmma_f32_16x16x4_f32(false, a, false, b,
                                                  (short)0, c, false, false);
    }

    // res_u,res_i are pre-scale: fold (1/(N_LAYERS+1))^2 = 1/16 into the score.
    const float sc = 1.0f / 16.0f;
    // Diagonal: m<8  -> VGPR m,   lane m
    //           m>=8 -> VGPR m-8, lane 16+m  (= 24 + (m-8))
    if (lane < 8) {
        int row = tbase + lane;
        if (row < nPred) obase[row] = c[lane] * sc;
    } else if (lane >= 24) {
        int row = tbase + lane - 16;
        if (row < nPred) obase[row] = c[lane - 24] * sc;
    }
}

// ---------------- host-side orchestration ----------------

extern "C" void kernel_launch(void* const* d_in, const int* in_sizes, int n_in,
                              void* d_out, int out_size, void* d_ws, size_t ws_size,
                              hipStream_t stream) {
    const float* emb_user  = (const float*)d_in[0];
    const float* emb_item  = (const float*)d_in[1];
    const int*   rate_u    = (const int*)d_in[2];
    const int*   rate_i    = (const int*)d_in[3];
    const int*   trust_src = (const int*)d_in[4];
    const int*   trust_dst = (const int*)d_in[5];
    const int*   pos_u     = (const int*)d_in[6];
    const int*   pos_i     = (const int*)d_in[7];
    const int*   neg_u     = (const int*)d_in[8];
    const int*   neg_i     = (const int*)d_in[9];

    const int nU = in_sizes[0] / EMBED;
    const int nI = in_sizes[1] / EMBED;
    const int E_rate  = in_sizes[2];
    const int E_trust = in_sizes[4];
    const int E_pred  = in_sizes[6];

    // Workspace layout (floats), each region padded to 64 floats (256 B)
    float* ws = (float*)d_ws;
    size_t off = 0;
    auto alloc = [&](size_t nfloats) -> float* {
        float* p = ws + off;
        off += (nfloats + 63) & ~(size_t)63;
        return p;
    };
    float* r_u  = alloc((size_t)nU);            // degree -> rsqrt norm (rate, user side)
    float* r_i  = alloc((size_t)nI);            // rate, item side
    float* r_ts = alloc((size_t)nU);            // trust src
    float* r_td = alloc((size_t)nU);            // trust dst
    size_t degRegion = (size_t)(r_td + (((size_t)nU + 63) & ~(size_t)63) - r_u);
    float* curU = alloc((size_t)nU * EMBED);
    float* newU = alloc((size_t)nU * EMBED);
    float* curI = alloc((size_t)nI * EMBED);
    float* newI = alloc((size_t)nI * EMBED);
    float* resU = alloc((size_t)nU * EMBED);
    float* resI = alloc((size_t)nI * EMBED);

    const int BS = 256;
    const long nU4 = (long)nU * EMBED4;
    const long nI4 = (long)nI * EMBED4;

    // 1) zero degree region, count degrees, convert to rsqrt
    {
        long n4 = (long)(degRegion / 4);
        fill_zero4_kernel<<<(int)((n4 + BS - 1) / BS), BS, 0, stream>>>((float4*)r_u, n4);
    }
    degree2_kernel<<<(E_rate + BS - 1) / BS, BS, 0, stream>>>(rate_u, rate_i, r_u, r_i, E_rate);
    degree2_kernel<<<(E_trust + BS - 1) / BS, BS, 0, stream>>>(trust_src, trust_dst, r_ts, r_td, E_trust);
    rsqrt_kernel<<<(int)((degRegion + BS - 1) / BS), BS, 0, stream>>>(r_u, (long)degRegion);

    // 2) cur = res = emb
    init_copy_kernel<<<(int)((nU4 + BS - 1) / BS), BS, 0, stream>>>(
        (const float4*)emb_user, (float4*)curU, (float4*)resU, nU4);
    init_copy_kernel<<<(int)((nI4 + BS - 1) / BS), BS, 0, stream>>>(
        (const float4*)emb_item, (float4*)curI, (float4*)resI, nI4);

    // 3) layers
    for (int layer = 0; layer < 3; ++layer) {
        fill_zero4_kernel<<<(int)((nU4 + BS - 1) / BS), BS, 0, stream>>>((float4*)newU, nU4);
        fill_zero4_kernel<<<(int)((nI4 + BS - 1) / BS), BS, 0, stream>>>((float4*)newI, nI4);

        long tRate = (long)E_rate * 16;
        conv_rate_kernel<<<(int)((tRate + BS - 1) / BS), BS, 0, stream>>>(
            (const float4*)curU, (const float4*)curI, newU, newI,
            rate_u, rate_i, r_u, r_i, E_rate);

        long tTrust = (long)E_trust * 16;
        conv_trust_kernel<<<(int)((tTrust + BS - 1) / BS), BS, 0, stream>>>(
            (const float4*)curU, newU, trust_src, trust_dst, r_ts, r_td, E_trust);

        accum_kernel<<<(int)((nU4 + BS - 1) / BS), BS, 0, stream>>>((float4*)resU, (const float4*)newU, nU4);
        accum_kernel<<<(int)((nI4 + BS - 1) / BS), BS, 0, stream>>>((float4*)resI, (const float4*)newI, nI4);

        // ping-pong cur <-> new
        float* tmp;
        tmp = curU; curU = newU; newU = tmp;
        tmp = curI; curI = newI; newI = tmp;
    }

    // 4) scoring: 16 edges per wave via V_WMMA_F32_16X16X4_F32
    int tilesPerSide = (E_pred + 15) / 16;
    int totalTiles = 2 * tilesPerSide;
    int wavesPerBlock = BS / 32;
    int scoreBlocks = (totalTiles + wavesPerBlock - 1) / wavesPerBlock;
    score_wmma_kernel<<<scoreBlocks, BS, 0, stream>>>(
        resU, resI, pos_u, pos_i, neg_u, neg_i,
        (float*)d_out, E_pred, tilesPerSide);

    (void)n_in; (void)out_size; (void)ws_size;
}